// TrajectoryPrototypeLayerMasked_20504173871794
// MI455X (gfx1250) — compile-verified
//
#include <hip/hip_runtime.h>
#include <hip/hip_bf16.h>

typedef __attribute__((ext_vector_type(2))) float v2f;
typedef __attribute__((ext_vector_type(8))) float v8f;

#define EPSV 1e-6f

// Problem constants (from reference setup)
#define NB 256
#define NS 128
#define ND 128
#define NP 32
#define NC 2
#define BSP (NB * NS * NP)   // 1048576

// ---------------------------------------------------------------------------
// Prep kernel: one block of 128 threads.
//  ws[0..127]    = mean_t[s]          (mean over batch of normalized_times)
//  ws[128..159]  = ||o_p||^2
//  ws[160..191]  = o_p . v_p
//  ws[192..223]  = ||v_p||^2
// ---------------------------------------------------------------------------
__global__ __launch_bounds__(128) void traj_prep_kernel(
    const float* __restrict__ ntimes,   // (B,S)
    const float* __restrict__ orig,     // (P,D)
    const float* __restrict__ vel,      // (P,D)
    float* __restrict__ ws)
{
    int tid = threadIdx.x;              // 0..127
    // mean over batch for segment s = tid
    float acc = 0.0f;
    for (int b = 0; b < NB; ++b)
        acc += ntimes[b * NS + tid];
    ws[tid] = acc * (1.0f / (float)NB);

    if (tid < NP) {
        const float* op = orig + tid * ND;
        const float* vp = vel + tid * ND;
        float so2 = 0.0f, sov = 0.0f, sv2 = 0.0f;
        for (int d = 0; d < ND; ++d) {
            float o = op[d], v = vp[d];
            so2 += o * o;
            sov += o * v;
            sv2 += v * v;
        }
        ws[128 + tid] = so2;
        ws[160 + tid] = sov;
        ws[192 + tid] = sv2;
    }
}

// ---------------------------------------------------------------------------
// Main kernel: grid = B blocks, 256 threads (8 wave32) per block.
// Block b covers rows r = b*S .. b*S+127 (all segments of one batch element).
// Wave w owns the 16-row tile starting at local row w*16 and computes
// dotO = z . origins^T and dotV = z . velocities^T with V_WMMA_F32_16X16X4_F32
// over K = D = 128 (32 K-steps), two N-tiles for P = 32.
// ---------------------------------------------------------------------------
__global__ __launch_bounds__(256) void traj_main_kernel(
    const float* __restrict__ z,        // (B,S,D)
    const float* __restrict__ wgt,      // (B,S)
    const float* __restrict__ orig,     // (P,D)
    const float* __restrict__ vel,      // (P,D)
    const float* __restrict__ ci,       // (P,C)
    const float* __restrict__ ws,       // prep results
    float* __restrict__ out)            // dist | sim | traj | logits
{
    __shared__ float mt[NS];        // mean_t per segment
    __shared__ float nw[NS];        // normalized weights for this b
    __shared__ float wl[NS];        // raw weights
    __shared__ float zn[NS];        // ||z||^2 per local row
    __shared__ float znh[256];      // half-row partials
    __shared__ float pterms[96];    // ||o||^2 | o.v | ||v||^2
    __shared__ float traj[NP];      // weighted similarity accumulator
    __shared__ float wsum_s;

    const int tid = threadIdx.x;
    const int b   = blockIdx.x;

    // ---- stage LDS: prep results, weights, traj init ----
    if (tid < NS)                  mt[tid] = ws[tid];
    if (tid >= 128 && tid < 224)   pterms[tid - 128] = ws[tid];
    if (tid < NS)                  wl[tid] = wgt[b * NS + tid];
    if (tid < NP)                  traj[tid] = 0.0f;

    // ---- per-row squared norms of z: two threads per row ----
    {
        int rl   = tid >> 1;           // local row 0..127
        int half = tid & 1;
        const float* zr = z + ((long)b * NS + rl) * ND + half * 64;
        float s = 0.0f;
        for (int d = 0; d < 64; ++d) s += zr[d] * zr[d];
        znh[tid] = s;
    }
    __syncthreads();

    if (tid == 0) {
        float s = 0.0f;
        for (int i = 0; i < NS; ++i) s += wl[i];
        wsum_s = fmaxf(s, EPSV);       // clip(sum, EPS)
    }
    if (tid < NS) zn[tid] = znh[2 * tid] + znh[2 * tid + 1];
    __syncthreads();
    if (tid < NS) nw[tid] = wl[tid] / wsum_s;
    __syncthreads();

    // ---- WMMA K-loop ----
    const int wave = tid >> 5;         // 0..7
    const int lane = tid & 31;
    const int m    = lane & 15;        // row-in-tile (A) / col (B,C,D)
    const int kh   = lane >> 4;        // K half: lanes 16-31 hold K=2,3
    const int rl0  = wave * 16;        // wave's first local row

    const float* za   = z    + ((long)b * NS + rl0 + m) * ND + 2 * kh;
    const float* bo0p = orig + (long)m        * ND + 2 * kh;
    const float* bo1p = orig + (long)(16 + m) * ND + 2 * kh;
    const float* bv0p = vel  + (long)m        * ND + 2 * kh;
    const float* bv1p = vel  + (long)(16 + m) * ND + 2 * kh;

    v8f accO0 = {}, accO1 = {}, accV0 = {}, accV1 = {};
    for (int k = 0; k < ND; k += 4) {
        v2f a   = *(const v2f*)(za   + k);
        v2f bo0 = *(const v2f*)(bo0p + k);
        v2f bo1 = *(const v2f*)(bo1p + k);
        v2f bv0 = *(const v2f*)(bv0p + k);
        v2f bv1 = *(const v2f*)(bv1p + k);
        accO0 = __builtin_amdgcn_wmma_f32_16x16x4_f32(false, a, false, bo0,
                                                      (short)0, accO0, false, false);
        accO1 = __builtin_amdgcn_wmma_f32_16x16x4_f32(false, a, false, bo1,
                                                      (short)0, accO1, false, false);
        accV0 = __builtin_amdgcn_wmma_f32_16x16x4_f32(false, a, false, bv0,
                                                      (short)0, accV0, false, false);
        accV1 = __builtin_amdgcn_wmma_f32_16x16x4_f32(false, a, false, bv1,
                                                      (short)0, accV1, false, false);
    }

    // ---- epilogue: dist, log-sim, weighted reduction ----
    // C/D layout: lane holds N = m, row = j + 8*kh for VGPR j.
    const int p0 = m;            // N-tile 0 prototype
    const int p1 = 16 + m;       // N-tile 1 prototype
    #pragma unroll
    for (int j = 0; j < 8; ++j) {
        const int rl = rl0 + j + 8 * kh;       // local row == segment s
        const float t  = mt[rl];
        const float zz = zn[rl];
        const float w  = nw[rl];
        const int rowo = (b * NS + rl) * NP;

        {   // tile 0
            float pn   = pterms[p0] + 2.0f * t * pterms[32 + p0]
                       + t * t * pterms[64 + p0];
            float dist = zz - 2.0f * accO0[j] - 2.0f * t * accV0[j] + pn;
            float sim  = logf(1.0f + 1.0f / (dist + EPSV));
            out[rowo + p0]       = dist;
            out[BSP + rowo + p0] = sim;
            atomicAdd(&traj[p0], sim * w);     // ds_add_f32
        }
        {   // tile 1
            float pn   = pterms[p1] + 2.0f * t * pterms[32 + p1]
                       + t * t * pterms[64 + p1];
            float dist = zz - 2.0f * accO1[j] - 2.0f * t * accV1[j] + pn;
            float sim  = logf(1.0f + 1.0f / (dist + EPSV));
            out[rowo + p1]       = dist;
            out[BSP + rowo + p1] = sim;
            atomicAdd(&traj[p1], sim * w);
        }
    }
    __syncthreads();

    // ---- trajectory similarities (B,P) and logits (B,C) ----
    if (tid < NP)
        out[2 * BSP + b * NP + tid] = traj[tid];
    if (tid < NC) {
        float acc = 0.0f;
        for (int p = 0; p < NP; ++p)
            acc += traj[p] * ci[p * NC + tid];
        out[2 * BSP + NB * NP + b * NC + tid] = acc;
    }
}

extern "C" void kernel_launch(void* const* d_in, const int* in_sizes, int n_in,
                              void* d_out, int out_size, void* d_ws, size_t ws_size,
                              hipStream_t stream) {
    const float* z      = (const float*)d_in[0];   // (B,S,D)
    const float* wgt    = (const float*)d_in[1];   // (B,S)
    const float* ntimes = (const float*)d_in[2];   // (B,S)
    const float* orig   = (const float*)d_in[3];   // (P,D)
    const float* vel    = (const float*)d_in[4];   // (P,D)
    const float* ci     = (const float*)d_in[5];   // (P,C)
    float* out = (float*)d_out;
    float* ws  = (float*)d_ws;

    traj_prep_kernel<<<1, 128, 0, stream>>>(ntimes, orig, vel, ws);
    traj_main_kernel<<<NB, 256, 0, stream>>>(z, wgt, orig, vel, ci, ws, out);
}